// MyModel_65429531787932
// MI455X (gfx1250) — compile-verified
//
#include <hip/hip_runtime.h>

// ---- problem sizes (match reference) ----
#define N0c 500000
#define N1c 200000
#define N2c 50000
#define E0c 2000000
#define E1c 750000
#define EPc 100000
#define D   128

typedef __attribute__((ext_vector_type(2))) float v2f;
typedef __attribute__((ext_vector_type(4))) float f4v;
typedef __attribute__((ext_vector_type(8))) float v8f;

#define LDS_STRIDE 132   // dwords; %4==0 (16B-aligned rows), %64!=0 (conflict-free b64 reads)

// -------- degree counting (segment_sum of ones) --------
__global__ void deg_kernel(const int* __restrict__ src, const int* __restrict__ dst, int nE,
                           unsigned* __restrict__ degS, unsigned* __restrict__ degD) {
    int i = blockIdx.x * blockDim.x + threadIdx.x;
    if (i < nE) {
        atomicAdd(&degS[src[i]], 1u);
        atomicAdd(&degD[dst[i]], 1u);
    }
}

// -------- SpMM scatter-add: one wave per edge, 128 floats via float4/lane --------
// If gidx != null:  row = feat[gidx[s]]  (fused embedding gather)
// If degS != null:  row scaled by rsqrt(max(deg,1)) (fused out-degree norm)
__global__ void spmm_kernel(const int* __restrict__ src, const int* __restrict__ dst, int nE,
                            const float* __restrict__ feat, const int* __restrict__ gidx,
                            const unsigned* __restrict__ degS, float* __restrict__ out) {
    int wid  = threadIdx.x >> 5;
    int lane = threadIdx.x & 31;
    int e = blockIdx.x * (blockDim.x >> 5) + wid;
    if (e >= nE) return;
    int s = src[e], d = dst[e];
    int rowIdx = gidx ? gidx[s] : s;
    const float* row = feat + (size_t)rowIdx * D;
    float scale = degS ? rsqrtf(fmaxf((float)degS[s], 1.0f)) : 1.0f;
    float4 v = ((const float4*)row)[lane];
    float* o = out + (size_t)d * D + (size_t)lane * 4;
    atomicAdd(o + 0, v.x * scale);
    atomicAdd(o + 1, v.y * scale);
    atomicAdd(o + 2, v.z * scale);
    atomicAdd(o + 3, v.w * scale);
}

// -------- dense [M,128] @ [128,128] via V_WMMA_F32_16X16X4_F32 --------
// Block = one M-tile (16 rows). 8 waves = 8 N-tiles. A tile (16x128 = 8KB) is
// staged into LDS once (non-temporal global reads -> WGP$ stays hot with W),
// then each wave runs 32 chained WMMAs reading A frags via ds_load_b64.
// Epilogue: relu(acc * rsqrt(in_deg) + bias) [* rsqrt(out_deg_next) if given]
__global__ void gemm_wmma_kernel(const float* __restrict__ A, const float* __restrict__ W,
                                 const float* __restrict__ bias,
                                 const unsigned* __restrict__ degIn,
                                 const unsigned* __restrict__ degOutNext,
                                 float* __restrict__ Hout) {
    __shared__ __align__(16) float As[16 * LDS_STRIDE];

    int tid = threadIdx.x;
    int mt  = blockIdx.x;

    // ---- cooperative stage of the 16x128 A tile (512 float4 by 256 threads) ----
    const f4v* Ag = (const f4v*)(A + (size_t)mt * 16 * D);
#pragma unroll
    for (int i = tid; i < 512; i += 256) {
        int row = i >> 5;     // 32 float4 per row
        int c4  = i & 31;
        f4v v = __builtin_nontemporal_load(&Ag[(size_t)row * 32 + c4]);
        *(f4v*)&As[row * LDS_STRIDE + c4 * 4] = v;
    }
    __syncthreads();

    int wid  = tid >> 5;
    int lane = tid & 31;
    int nt   = wid;                       // 8 waves -> 8 column tiles
    int row_base = mt * 16;
    int kh   = (lane >> 4) * 2;           // K sub-offset: 0 (lanes 0-15) or 2 (lanes 16-31)
    int col  = nt * 16 + (lane & 15);

    const float* as   = As + (lane & 15) * LDS_STRIDE + kh;  // A[m][k+kh] in LDS
    const float* wcol = W + (size_t)kh * D + col;            // W[k+kh][col]

    v8f c = {0.0f, 0.0f, 0.0f, 0.0f, 0.0f, 0.0f, 0.0f, 0.0f};
#pragma unroll
    for (int k = 0; k < D; k += 4) {
        float2 av = *(const float2*)(as + k);                // ds_load_b64, conflict-free
        v2f a, b;
        a.x = av.x;
        a.y = av.y;
        b.x = wcol[(size_t)k * D];
        b.y = wcol[(size_t)(k + 1) * D];
        c = __builtin_amdgcn_wmma_f32_16x16x4_f32(false, a, false, b, (short)0, c,
                                                  false, false);
    }

    float bb = bias[col];
    int rbase = row_base + ((lane >> 4) * 8);   // lanes 0-15 -> rows g, lanes 16-31 -> g+8
#pragma unroll
    for (int g = 0; g < 8; ++g) {
        int row = rbase + g;
        float rsin = rsqrtf(fmaxf((float)degIn[row], 1.0f));
        float v = fmaxf(c[g] * rsin + bb, 0.0f);
        if (degOutNext) v *= rsqrtf(fmaxf((float)degOutNext[row], 1.0f));
        Hout[(size_t)row * D + col] = v;
    }
}

// -------- edge predictor: [h[src] ; h[dst]] @ Wp(256x2) + bp, one wave/edge --------
__global__ void edge_pred_kernel(const int* __restrict__ esrc, const int* __restrict__ edst,
                                 const float* __restrict__ H, const float* __restrict__ Wp,
                                 const float* __restrict__ bp, float* __restrict__ out, int nE) {
    int wid  = threadIdx.x >> 5;
    int lane = threadIdx.x & 31;
    int e = blockIdx.x * (blockDim.x >> 5) + wid;
    if (e >= nE) return;
    int s = esrc[e], d = edst[e];
    float4 hs = ((const float4*)(H + (size_t)s * D))[lane];
    float4 hd = ((const float4*)(H + (size_t)d * D))[lane];
    float hsv[4] = {hs.x, hs.y, hs.z, hs.w};
    float hdv[4] = {hd.x, hd.y, hd.z, hd.w};
    const float* wpS = Wp + (size_t)(4 * lane) * 2;         // rows 4*lane..4*lane+3
    const float* wpD = Wp + (size_t)(128 + 4 * lane) * 2;   // rows 128+4*lane..
    float acc0 = 0.0f, acc1 = 0.0f;
#pragma unroll
    for (int j = 0; j < 4; ++j) {
        acc0 += hsv[j] * wpS[2 * j]     + hdv[j] * wpD[2 * j];
        acc1 += hsv[j] * wpS[2 * j + 1] + hdv[j] * wpD[2 * j + 1];
    }
#pragma unroll
    for (int off = 16; off >= 1; off >>= 1) {
        acc0 += __shfl_xor(acc0, off, 32);
        acc1 += __shfl_xor(acc1, off, 32);
    }
    if (lane == 0) {
        out[(size_t)e * 2 + 0] = acc0 + bp[0];
        out[(size_t)e * 2 + 1] = acc1 + bp[1];
    }
}

extern "C" void kernel_launch(void* const* d_in, const int* in_sizes, int n_in,
                              void* d_out, int out_size, void* d_ws, size_t ws_size,
                              hipStream_t stream) {
    const float* emb = (const float*)d_in[0];
    const float* W1  = (const float*)d_in[1];
    const float* b1  = (const float*)d_in[2];
    const float* W2  = (const float*)d_in[3];
    const float* b2  = (const float*)d_in[4];
    const float* Wp  = (const float*)d_in[5];
    const float* bp  = (const float*)d_in[6];
    const int* input_nodes = (const int*)d_in[7];
    const int* b0_src = (const int*)d_in[8];
    const int* b0_dst = (const int*)d_in[9];
    const int* b1_src = (const int*)d_in[10];
    const int* b1_dst = (const int*)d_in[11];
    const int* e_src  = (const int*)d_in[12];
    const int* e_dst  = (const int*)d_in[13];
    float* out = (float*)d_out;

    // ---- workspace layout ----
    char* ws = (char*)d_ws;
    unsigned* deg0_src = (unsigned*)ws;            // N0
    unsigned* deg0_dst = deg0_src + N0c;           // N1
    unsigned* deg1_src = deg0_dst + N1c;           // N1
    unsigned* deg1_dst = deg1_src + N1c;           // N2
    size_t degBytes = (size_t)(N0c + 2 * N1c + N2c) * sizeof(unsigned);
    size_t off = (degBytes + 1023) & ~(size_t)1023;
    float* agg1 = (float*)(ws + off);              // N1*128 floats (102.4 MB)
    float* h1   = agg1 + (size_t)N1c * D;          // N1*128 floats (102.4 MB)
    float* agg2 = agg1;                            // reuse (agg1 dead after GEMM1)
    float* h2   = agg1 + (size_t)N2c * D;          // reuse (fits inside agg1 region)

    // zero degree counters and agg1
    hipMemsetAsync(ws, 0, off, stream);
    hipMemsetAsync(agg1, 0, (size_t)N1c * D * sizeof(float), stream);

    // degrees for both blocks
    deg_kernel<<<(E0c + 255) / 256, 256, 0, stream>>>(b0_src, b0_dst, E0c, deg0_src, deg0_dst);
    deg_kernel<<<(E1c + 255) / 256, 256, 0, stream>>>(b1_src, b1_dst, E1c, deg1_src, deg1_dst);

    // layer 1 SpMM: agg1[d] += emb[input_nodes[s]] * rsqrt(out_deg0[s])
    spmm_kernel<<<(E0c + 7) / 8, 256, 0, stream>>>(b0_src, b0_dst, E0c,
                                                   emb, input_nodes, deg0_src, agg1);
    // layer 1 GEMM (WMMA fp32): h1 = relu((agg1*rs_in)@W1 + b1) * rsqrt(out_deg1)
    gemm_wmma_kernel<<<N1c / 16, 256, 0, stream>>>(agg1, W1, b1, deg0_dst, deg1_src, h1);

    // layer 2 SpMM: agg2[d] += h1[s]  (out-degree scale pre-folded into h1)
    hipMemsetAsync(agg2, 0, (size_t)N2c * D * sizeof(float), stream);
    spmm_kernel<<<(E1c + 7) / 8, 256, 0, stream>>>(b1_src, b1_dst, E1c,
                                                   h1, nullptr, nullptr, agg2);
    // layer 2 GEMM (WMMA fp32): h2 = relu((agg2*rs_in)@W2 + b2)
    gemm_wmma_kernel<<<N2c / 16, 256, 0, stream>>>(agg2, W2, b2, deg1_dst, nullptr, h2);

    // edge prediction
    edge_pred_kernel<<<(EPc + 7) / 8, 256, 0, stream>>>(e_src, e_dst, h2, Wp, bp, out, EPc);
}